// MultiAssetRiskModel_58042188038522
// MI455X (gfx1250) — compile-verified
//
#include <hip/hip_runtime.h>
#include <hip/hip_bf16.h>

typedef __attribute__((ext_vector_type(2))) float v2f;
typedef __attribute__((ext_vector_type(8))) float v8f;

#define N_ASSETS 4096
#define T_HIST   252
#define CORR_W   60
#define REG_W    30
#define TW       64      // padded time window (60 -> 64)
#define B_PORT   1024
#define TARGET_VOL 0.0094491117f  // 0.15f / sqrtf(252.0f)

// ---------------- ws layout (float offsets) ----------------
#define WS_XC       0                         // 64*4096 = 262144
#define WS_Y        (WS_XC + TW * N_ASSETS)   // 1024*64 = 65536
#define WS_P_SUM    (WS_Y + B_PORT * TW)      // 256 (unused result)
#define WS_P_SSQ    (WS_P_SUM + 256)          // 256
#define WS_R30_SUM  (WS_P_SSQ + 256)          // 32
#define WS_R30_SSQ  (WS_R30_SUM + 32)         // 32
#define WS_H_SUM    (WS_R30_SSQ + 32)         // 128
#define WS_H_SSQ    (WS_H_SUM + 128)          // 128
#define WS_VOLPEN   (WS_H_SSQ + 128)          // 1
#define WS_VAR5     (WS_VOLPEN + 1)           // 1

// ------------------------------------------------------------------
// K1: column means over last 60 rows, write centered Xc padded to 64 rows
// ------------------------------------------------------------------
__global__ void prep_xc(const float* __restrict__ R, float* __restrict__ Xc) {
    int j = blockIdx.x * blockDim.x + threadIdx.x;           // asset 0..4095
    const float* col = R + (size_t)(T_HIST - CORR_W) * N_ASSETS + j;
    float s = 0.0f;
    #pragma unroll 4
    for (int t = 0; t < CORR_W; ++t) s += col[(size_t)t * N_ASSETS];
    float mean = s * (1.0f / (float)CORR_W);
    #pragma unroll 4
    for (int t = 0; t < CORR_W; ++t)
        Xc[(size_t)t * N_ASSETS + j] = col[(size_t)t * N_ASSETS] - mean;
    for (int t = CORR_W; t < TW; ++t)
        Xc[(size_t)t * N_ASSETS + j] = 0.0f;
}

// ------------------------------------------------------------------
// K2: Y = P @ Xc^T  (M=1024, N=64, K=4096) via V_WMMA_F32_16X16X4_F32.
// One wave handles a 16-row M block and all 64 N columns (4 accumulators).
// A layout: lanes 0-15 -> M=lane, VGPR{0,1}=K{0,1}; lanes 16-31 -> K{2,3}.
// B layout mirrors A with N in place of M. D: VGPR r -> M=r+8*half, lane&15 -> N.
// ------------------------------------------------------------------
__global__ __launch_bounds__(128)
void gemm_wmma(const float* __restrict__ P, const float* __restrict__ Xc,
               float* __restrict__ Y) {
    const int wave = threadIdx.x >> 5;                 // 0..3
    const int lane = threadIdx.x & 31;
    const int mblk = blockIdx.x * 4 + wave;            // 0..63
    const int half = lane >> 4;                        // 0 or 1 -> K pair
    const int l15  = lane & 15;

    const float* aPtr  = P  + (size_t)(mblk * 16 + l15) * N_ASSETS + half * 2;
    const float* b0Ptr = Xc + (size_t)( 0 + l15) * N_ASSETS + half * 2;
    const float* b1Ptr = Xc + (size_t)(16 + l15) * N_ASSETS + half * 2;
    const float* b2Ptr = Xc + (size_t)(32 + l15) * N_ASSETS + half * 2;
    const float* b3Ptr = Xc + (size_t)(48 + l15) * N_ASSETS + half * 2;

    v8f acc0 = {}; v8f acc1 = {}; v8f acc2 = {}; v8f acc3 = {};

    #pragma unroll 2
    for (int k = 0; k < N_ASSETS; k += 4) {
        v2f a  = *(const v2f*)(aPtr  + k);
        v2f b0 = *(const v2f*)(b0Ptr + k);
        v2f b1 = *(const v2f*)(b1Ptr + k);
        v2f b2 = *(const v2f*)(b2Ptr + k);
        v2f b3 = *(const v2f*)(b3Ptr + k);
        __builtin_prefetch(aPtr + k + 512, 0, 0);      // global_prefetch_b8
        acc0 = __builtin_amdgcn_wmma_f32_16x16x4_f32(false, a, false, b0,
                                                     (short)0, acc0, false, false);
        acc1 = __builtin_amdgcn_wmma_f32_16x16x4_f32(false, a, false, b1,
                                                     (short)0, acc1, false, false);
        acc2 = __builtin_amdgcn_wmma_f32_16x16x4_f32(false, a, false, b2,
                                                     (short)0, acc2, false, false);
        acc3 = __builtin_amdgcn_wmma_f32_16x16x4_f32(false, a, false, b3,
                                                     (short)0, acc3, false, false);
    }

    #pragma unroll
    for (int r = 0; r < 8; ++r) {
        int ro = mblk * 16 + r + 8 * half;
        Y[(size_t)ro * TW +  0 + l15] = acc0[r];
        Y[(size_t)ro * TW + 16 + l15] = acc1[r];
        Y[(size_t)ro * TW + 32 + l15] = acc2[r];
        Y[(size_t)ro * TW + 48 + l15] = acc3[r];
    }
}

// ------------------------------------------------------------------
// K3: variance[b] = ||Y[b,:]||^2 / 59; volatility; vol-penalty partial sum
// ------------------------------------------------------------------
__global__ void variance_kernel(const float* __restrict__ Y,
                                float* __restrict__ out_var,
                                float* __restrict__ out_vol,
                                float* __restrict__ volpen_out) {
    __shared__ float sh[B_PORT];
    int b = threadIdx.x;
    float ss = 0.0f;
    #pragma unroll 8
    for (int t = 0; t < TW; ++t) { float y = Y[(size_t)b * TW + t]; ss += y * y; }
    float var = ss * (1.0f / 59.0f);
    out_var[b] = var;
    out_vol[b] = sqrtf(var + 1e-8f);
    float pv = sqrtf(var) - TARGET_VOL;
    sh[b] = pv > 0.0f ? pv : 0.0f;
    __syncthreads();
    for (int s = B_PORT / 2; s > 0; s >>= 1) {
        if (b < s) sh[b] += sh[b + s];
        __syncthreads();
    }
    if (b == 0) volpen_out[0] = sh[0];
}

// ------------------------------------------------------------------
// K4/K5: grid-stride sum + sumsq partials (deterministic tree reduce)
// ------------------------------------------------------------------
__global__ void partial_sums(const float* __restrict__ x, int n,
                             float* __restrict__ psum, float* __restrict__ pssq) {
    __shared__ float s1[256];
    __shared__ float s2[256];
    int tid = threadIdx.x;
    float a = 0.0f, b = 0.0f;
    for (int i = blockIdx.x * blockDim.x + tid; i < n; i += gridDim.x * blockDim.x) {
        float v = x[i];
        a += v; b += v * v;
    }
    s1[tid] = a; s2[tid] = b;
    __syncthreads();
    for (int s = 128; s > 0; s >>= 1) {
        if (tid < s) { s1[tid] += s1[tid + s]; s2[tid] += s2[tid + s]; }
        __syncthreads();
    }
    if (tid == 0) { psum[blockIdx.x] = s1[0]; pssq[blockIdx.x] = s2[0]; }
}

// ------------------------------------------------------------------
// K6: 5% VaR = 51st smallest of 1024 returns (rank selection in LDS)
// ------------------------------------------------------------------
__global__ void var_select(const float* __restrict__ pr, float* __restrict__ var5) {
    __shared__ float sh[B_PORT];
    int i = threadIdx.x;
    sh[i] = pr[i];
    __syncthreads();
    float v = sh[i];
    int rank = 0;
    for (int j = 0; j < B_PORT; ++j) {
        float u = sh[j];
        rank += (u < v) || (u == v && j < i);
    }
    if (rank == 51) *var5 = v;
}

// ------------------------------------------------------------------
// K7: reduce all partials, compute scalar penalties
// ------------------------------------------------------------------
__device__ float block_reduce_256(float v, float* sh) {
    int t = threadIdx.x;
    sh[t] = v;
    __syncthreads();
    for (int s = 128; s > 0; s >>= 1) {
        if (t < s) sh[t] += sh[t + s];
        __syncthreads();
    }
    float r = sh[0];
    __syncthreads();
    return r;
}

__global__ void finalize(const float* __restrict__ ws, float* __restrict__ out) {
    __shared__ float sh[256];
    int t = threadIdx.x;

    float p_ssq  = block_reduce_256(ws[WS_P_SSQ + t], sh);
    float r_sum  = block_reduce_256(t < 32  ? ws[WS_R30_SUM + t] : 0.0f, sh);
    float r_ssq  = block_reduce_256(t < 32  ? ws[WS_R30_SSQ + t] : 0.0f, sh);
    float h_sum  = block_reduce_256(t < 128 ? ws[WS_H_SUM  + t] : 0.0f, sh);
    float h_ssq  = block_reduce_256(t < 128 ? ws[WS_H_SSQ  + t] : 0.0f, sh);

    if (t == 0) {
        // concentration
        float hhi  = p_ssq * (1.0f / (float)B_PORT);
        float conc = fmaxf(hhi - 0.01f, 0.0f) * 15.0f;
        // regime vols (flattened std, ddof=1)
        const float n30 = (float)(REG_W * N_ASSETS);
        const float nh  = (float)((T_HIST - REG_W) * N_ASSETS);
        float rvol = sqrtf(fmaxf((r_ssq - r_sum * r_sum / n30) / (n30 - 1.0f), 0.0f));
        float hvol = sqrtf(fmaxf((h_ssq - h_sum * h_sum / nh ) / (nh  - 1.0f), 0.0f));
        float vrc  = fabsf(rvol - hvol) / (hvol + 1e-8f);
        float regime = vrc * 0.1f;
        // VaR penalty
        float var5   = ws[WS_VAR5];
        float varpen = fmaxf(-var5 - 0.02f, 0.0f) * 10.0f;
        // vol penalty (mean * 5)
        float volpen = ws[WS_VOLPEN] * (5.0f / (float)B_PORT);

        out[0]    = varpen + conc + volpen + regime;  // total_risk_penalty
        out[2049] = varpen;
        out[2050] = conc;
        out[2051] = volpen;
        out[2052] = regime;
        out[2053] = vrc;
    }
}

// ------------------------------------------------------------------
extern "C" void kernel_launch(void* const* d_in, const int* in_sizes, int n_in,
                              void* d_out, int out_size, void* d_ws, size_t ws_size,
                              hipStream_t stream) {
    const float* positions = (const float*)d_in[0];   // 1024*4096
    const float* returns   = (const float*)d_in[1];   // 252*4096
    const float* pret      = (const float*)d_in[2];   // 1024
    float* out = (float*)d_out;
    float* ws  = (float*)d_ws;

    float* Xc = ws + WS_XC;
    float* Y  = ws + WS_Y;

    // 1) centered recent window (padded 64 rows)
    prep_xc<<<N_ASSETS / 256, 256, 0, stream>>>(returns, Xc);

    // 2) Y = P @ Xc^T via f32 WMMA (64 M-blocks, 4 waves/block)
    gemm_wmma<<<16, 128, 0, stream>>>(positions, Xc, Y);

    // 3) per-portfolio variance / volatility / vol-penalty partial
    variance_kernel<<<1, B_PORT, 0, stream>>>(Y, out + 1, out + 1 + B_PORT,
                                              ws + WS_VOLPEN);

    // 4) HHI: sum of squares of all positions (sum result unused)
    partial_sums<<<256, 256, 0, stream>>>(positions, B_PORT * N_ASSETS,
                                          ws + WS_P_SUM, ws + WS_P_SSQ);

    // 5) regime windows: recent 30 rows and historical 222 rows (flattened)
    partial_sums<<<32, 256, 0, stream>>>(returns + (size_t)(T_HIST - REG_W) * N_ASSETS,
                                         REG_W * N_ASSETS,
                                         ws + WS_R30_SUM, ws + WS_R30_SSQ);
    partial_sums<<<128, 256, 0, stream>>>(returns, (T_HIST - REG_W) * N_ASSETS,
                                          ws + WS_H_SUM, ws + WS_H_SSQ);

    // 6) 5% VaR order statistic
    var_select<<<1, B_PORT, 0, stream>>>(pret, ws + WS_VAR5);

    // 7) scalars
    finalize<<<1, 256, 0, stream>>>(ws, out);
}